// GATGNN_54881092108710
// MI455X (gfx1250) — compile-verified
//
#include <hip/hip_runtime.h>

// =====================  GAT-GNN forward for MI455X (gfx1250)  =====================
// Strategy: split concat-GEMM into node-level (x@Wx) + edge-level (ea@We) parts,
// run all GEMMs through V_WMMA_F32_16X16X4_F32 (wave32, 16-row tiles x 64 cols),
// segment-softmax via bit-pattern atomicMax + exp/denom, scatter-mean via f32 atomics.

typedef __attribute__((ext_vector_type(2))) float v2f;
typedef __attribute__((ext_vector_type(8))) float v8f;

namespace {
constexpr int N_NODES  = 50000;
constexpr int N_EDGES  = 1600000;
constexpr int N_GRAPHS = 500;
constexpr int DIM      = 64;
constexpr int N_FEAT   = 100;
constexpr int E_FEAT   = 50;
constexpr int N_LAYERS = 3;
constexpr int WPB = 8;          // waves per block
constexpr int TPB = WPB * 32;   // 256 threads (8 wave32)
}

#define DEVI __device__ __forceinline__

DEVI float lrelu02(float v)   { return v >= 0.f ? v : 0.2f * v; }
DEVI float softplusf(float x) { return fmaxf(x, 0.f) + log1pf(expf(-fabsf(x))); }

// One wave computes a 16xK (A, row-major lda) x Kx64 (B, row-major 64) tile into c[4].
// VGPR layouts per CDNA5 ISA 7.12.2 (32-bit A 16x4 / B 4x16 / C 16x16).
template<int K, bool LEAKY>
DEVI void wmma_16xK_x64(const float* __restrict__ A, int lda,
                        const float* __restrict__ B, v8f c[4]) {
  const int lane = threadIdx.x & 31;
  const int l16  = lane & 15;
  const int half = lane >> 4;
  #pragma unroll
  for (int k = 0; k < K; k += 4) {
    const int kk = k + 2 * half;
    v2f a;
    a.x = (kk     < K) ? A[(size_t)l16 * lda + kk]     : 0.f;
    a.y = (kk + 1 < K) ? A[(size_t)l16 * lda + kk + 1] : 0.f;
    if (LEAKY) { a.x = lrelu02(a.x); a.y = lrelu02(a.y); }
    #pragma unroll
    for (int t = 0; t < 4; ++t) {
      const int n = t * 16 + l16;
      v2f b;
      b.x = (kk     < K) ? B[(size_t)kk * 64 + n]       : 0.f;
      b.y = (kk + 1 < K) ? B[(size_t)(kk + 1) * 64 + n] : 0.f;
      c[t] = __builtin_amdgcn_wmma_f32_16x16x4_f32(false, a, false, b,
                                                   (short)0, c[t], false, false);
    }
  }
}

DEVI void zero_c(v8f c[4]) {
  #pragma unroll
  for (int t = 0; t < 4; ++t)
    c[t] = v8f{0.f, 0.f, 0.f, 0.f, 0.f, 0.f, 0.f, 0.f};
}

// ---------- pre FC: h = relu(x @ pre_w + pre_b), K = 100 ----------
__global__ __launch_bounds__(TPB) void pre_fc_kernel(
    const float* __restrict__ X, const float* __restrict__ W,
    const float* __restrict__ bias, float* __restrict__ out, int mtiles) {
  const int wid = blockIdx.x * WPB + (threadIdx.x >> 5);
  if (wid >= mtiles) return;
  const int m0 = wid * 16;
  v8f c[4]; zero_c(c);
  wmma_16xK_x64<N_FEAT, false>(X + (size_t)m0 * N_FEAT, N_FEAT, W, c);
  const int lane = threadIdx.x & 31, l16 = lane & 15, half = lane >> 4;
  #pragma unroll
  for (int t = 0; t < 4; ++t) {
    const int n = t * 16 + l16;
    const float bn = bias[n];
    #pragma unroll
    for (int j = 0; j < 8; ++j) {
      const int m = m0 + half * 8 + j;
      out[(size_t)m * 64 + n] = fmaxf(c[t][j] + bn, 0.f);
    }
  }
}

// ---------- node matmul: xW = h @ Wx, K = 64, no epilogue ----------
__global__ __launch_bounds__(TPB) void xw_kernel(
    const float* __restrict__ H, const float* __restrict__ Wx,
    float* __restrict__ out, int mtiles) {
  const int wid = blockIdx.x * WPB + (threadIdx.x >> 5);
  if (wid >= mtiles) return;
  const int m0 = wid * 16;
  v8f c[4]; zero_c(c);
  wmma_16xK_x64<DIM, false>(H + (size_t)m0 * DIM, DIM, Wx, c);
  const int lane = threadIdx.x & 31, l16 = lane & 15, half = lane >> 4;
  #pragma unroll
  for (int t = 0; t < 4; ++t) {
    const int n = t * 16 + l16;
    #pragma unroll
    for (int j = 0; j < 8; ++j) {
      const int m = m0 + half * 8 + j;
      out[(size_t)m * 64 + n] = c[t][j];
    }
  }
}

// ---------- edge pass 1: eW = lrelu(ea)@We; alpha logits + segment max ----------
__global__ __launch_bounds__(TPB) void edge_attn_kernel(
    const float* __restrict__ ea, const float* __restrict__ We,
    const float* __restrict__ xW, const int* __restrict__ src,
    const int* __restrict__ dst, const float* __restrict__ att,
    float* __restrict__ alphaOut, int* __restrict__ nodeMaxBits, int etiles) {
  const int wid = blockIdx.x * WPB + (threadIdx.x >> 5);
  if (wid >= etiles) return;
  const int e0 = wid * 16;
  v8f c[4]; zero_c(c);
  wmma_16xK_x64<E_FEAT, true>(ea + (size_t)e0 * E_FEAT, E_FEAT, We, c);
  const int lane = threadIdx.x & 31, l16 = lane & 15, half = lane >> 4;
  float ah[4], aj[4];
  #pragma unroll
  for (int t = 0; t < 4; ++t) {
    ah[t] = att[t * 16 + l16];        // pairs with hi (dst side)
    aj[t] = att[64 + t * 16 + l16];   // pairs with hj (src side)
  }
  float red[8];
  #pragma unroll
  for (int j = 0; j < 8; ++j) {
    const int m = e0 + half * 8 + j;
    const int d = dst[m], s = src[m];
    float acc = 0.f;
    #pragma unroll
    for (int t = 0; t < 4; ++t) {
      const int n = t * 16 + l16;
      const float ew = c[t][j];
      acc += softplusf(xW[(size_t)d * 64 + n] + ew) * ah[t]
           + softplusf(xW[(size_t)s * 64 + n] + ew) * aj[t];
    }
    // reduce over the 16 lanes of this half (masks 1..8 stay within the half)
    acc += __shfl_xor(acc, 1, 32);
    acc += __shfl_xor(acc, 2, 32);
    acc += __shfl_xor(acc, 4, 32);
    acc += __shfl_xor(acc, 8, 32);
    red[j] = acc;
  }
  if (l16 == 0) {
    #pragma unroll
    for (int j = 0; j < 8; ++j) {
      const int m = e0 + half * 8 + j;
      const float alpha = softplusf(red[j]);   // > 0 always
      alphaOut[m] = alpha;
      // non-negative floats: int-bit compare == float compare
      atomicMax(&nodeMaxBits[dst[m]], __float_as_int(alpha));
    }
  }
}

// ---------- edge pass 2: e = exp(alpha - max[dst]); denom += e ----------
__global__ void edge_exp_kernel(const float* __restrict__ alpha,
                                const int* __restrict__ dst,
                                const int* __restrict__ nodeMaxBits,
                                float* __restrict__ eexp,
                                float* __restrict__ denom, int nE) {
  const int i = blockIdx.x * blockDim.x + threadIdx.x;
  if (i >= nE) return;
  const int d = dst[i];
  const float e = expf(alpha[i] - __int_as_float(nodeMaxBits[d]));
  eexp[i] = e;
  atomicAdd(&denom[d], e);
}

// ---------- edge pass 3: recompute eW/hj, scatter hj * softmax into accum ----------
__global__ __launch_bounds__(TPB) void edge_scatter_kernel(
    const float* __restrict__ ea, const float* __restrict__ We,
    const float* __restrict__ xW, const int* __restrict__ src,
    const int* __restrict__ dst, const float* __restrict__ eexp,
    const float* __restrict__ denom, float* __restrict__ accum, int etiles) {
  const int wid = blockIdx.x * WPB + (threadIdx.x >> 5);
  if (wid >= etiles) return;
  const int e0 = wid * 16;
  v8f c[4]; zero_c(c);
  wmma_16xK_x64<E_FEAT, true>(ea + (size_t)e0 * E_FEAT, E_FEAT, We, c);
  const int lane = threadIdx.x & 31, l16 = lane & 15, half = lane >> 4;
  #pragma unroll
  for (int j = 0; j < 8; ++j) {
    const int m = e0 + half * 8 + j;
    const int s = src[m], d = dst[m];
    const float w = eexp[m] / (denom[d] + 1e-16f);
    #pragma unroll
    for (int t = 0; t < 4; ++t) {
      const int n = t * 16 + l16;
      const float hj = softplusf(xW[(size_t)s * 64 + n] + c[t][j]);
      atomicAdd(&accum[(size_t)d * 64 + n], hj * w);
    }
  }
}

// ---------- small helper kernels ----------
__global__ void degree_kernel(const int* __restrict__ dst, float* __restrict__ cnt, int nE) {
  const int i = blockIdx.x * blockDim.x + threadIdx.x;
  if (i < nE) atomicAdd(&cnt[dst[i]], 1.f);
}

__global__ void gcount_kernel(const int* __restrict__ batch, float* __restrict__ gcnt, int n) {
  const int i = blockIdx.x * blockDim.x + threadIdx.x;
  if (i < n) atomicAdd(&gcnt[batch[i]], 1.f);
}

// accum := accum / max(cnt,1) + bias   (scatter-mean + conv bias, in place)
__global__ void mean_bias_kernel(float* __restrict__ accum, const float* __restrict__ cnt,
                                 const float* __restrict__ bias, int n) {
  const size_t idx = (size_t)blockIdx.x * blockDim.x + threadIdx.x;
  if (idx >= (size_t)n * 64) return;
  const int row = (int)(idx >> 6), d = (int)(idx & 63);
  accum[idx] = accum[idx] / fmaxf(cnt[row], 1.f) + bias[d];
}

// per-dim sum & sumsq over nodes -> sums[0:64]=sum, sums[64:128]=sumsq
__global__ void bn_stats_kernel(const float* __restrict__ v, float* __restrict__ sums, int n) {
  __shared__ float ls[128];
  if (threadIdx.x < 128) ls[threadIdx.x] = 0.f;
  __syncthreads();
  const int d = threadIdx.x & 63;
  const int rl = threadIdx.x >> 6;  // 0..3
  float s = 0.f, ss = 0.f;
  for (int r = blockIdx.x * 4 + rl; r < n; r += gridDim.x * 4) {
    const float x = v[(size_t)r * 64 + d];
    s += x; ss += x * x;
  }
  atomicAdd(&ls[d], s);
  atomicAdd(&ls[64 + d], ss);
  __syncthreads();
  if (threadIdx.x < 128) atomicAdd(&sums[threadIdx.x], ls[threadIdx.x]);
}

__global__ void bn_apply_kernel(const float* __restrict__ v, const float* __restrict__ sums,
                                const float* __restrict__ g, const float* __restrict__ b,
                                float* __restrict__ out, int n) {
  const size_t idx = (size_t)blockIdx.x * blockDim.x + threadIdx.x;
  if (idx >= (size_t)n * 64) return;
  const int d = (int)(idx & 63);
  const float mu  = sums[d] / (float)n;
  const float var = sums[64 + d] / (float)n - mu * mu;
  out[idx] = (v[idx] - mu) * rsqrtf(var + 1e-5f) * g[d] + b[d];
}

__global__ void pool_kernel(const float* __restrict__ h, const int* __restrict__ batch,
                            float* __restrict__ pooled, int n) {
  const size_t idx = (size_t)blockIdx.x * blockDim.x + threadIdx.x;
  if (idx >= (size_t)n * 64) return;
  const int row = (int)(idx >> 6), d = (int)(idx & 63);
  atomicAdd(&pooled[(size_t)batch[row] * 64 + d], h[idx]);
}

// per graph: mean-pool -> relu(@post_w + post_b) -> @out_w + out_b
__global__ void head_kernel(const float* __restrict__ pooled, const float* __restrict__ gcnt,
                            const float* __restrict__ postW, const float* __restrict__ postB,
                            const float* __restrict__ outW, const float* __restrict__ outB,
                            float* __restrict__ y) {
  const int g = blockIdx.x;
  const int o = threadIdx.x;  // 64 threads
  __shared__ float pm[64];
  __shared__ float red[64];
  const float invc = 1.f / fmaxf(gcnt[g], 1.f);
  pm[o] = pooled[(size_t)g * 64 + o] * invc;
  __syncthreads();
  float acc = postB[o];
  #pragma unroll
  for (int d = 0; d < 64; ++d) acc += pm[d] * postW[d * 64 + o];
  red[o] = fmaxf(acc, 0.f) * outW[o];
  __syncthreads();
  for (int st = 32; st > 0; st >>= 1) {
    if (o < st) red[o] += red[o + st];
    __syncthreads();
  }
  if (o == 0) y[g] = red[0] + outB[0];
}

extern "C" void kernel_launch(void* const* d_in, const int* in_sizes, int n_in,
                              void* d_out, int out_size, void* d_ws, size_t ws_size,
                              hipStream_t stream) {
  (void)in_sizes; (void)n_in; (void)out_size; (void)ws_size;
  const float* x         = (const float*)d_in[0];
  const float* edge_attr = (const float*)d_in[1];
  const int*   edge_idx  = (const int*)  d_in[2];
  const int*   batch     = (const int*)  d_in[3];
  const float* pre_w     = (const float*)d_in[4];
  const float* pre_b     = (const float*)d_in[5];
  const float* conv_W    = (const float*)d_in[6];   // [3,114,64]
  const float* conv_att  = (const float*)d_in[7];   // [3,1,128]
  const float* conv_bias = (const float*)d_in[8];   // [3,64]
  const float* bn_g      = (const float*)d_in[9];
  const float* bn_b      = (const float*)d_in[10];
  const float* post_w    = (const float*)d_in[11];
  const float* post_b    = (const float*)d_in[12];
  const float* out_w     = (const float*)d_in[13];
  const float* out_b     = (const float*)d_in[14];

  const int* srcIdx = edge_idx;            // edge_index[0]
  const int* dstIdx = edge_idx + N_EDGES;  // edge_index[1]

  // -------- workspace layout (~52 MB of floats) --------
  float* ws     = (float*)d_ws;
  float* h      = ws;                               // N*64
  float* xW     = h      + (size_t)N_NODES * 64;    // N*64
  float* accum  = xW     + (size_t)N_NODES * 64;    // N*64
  float* alpha  = accum  + (size_t)N_NODES * 64;    // E
  float* eexp   = alpha  + (size_t)N_EDGES;         // E
  float* denom  = eexp   + (size_t)N_EDGES;         // N
  float* cnt    = denom  + (size_t)N_NODES;         // N (must be contiguous with gcnt)
  float* gcnt   = cnt    + (size_t)N_NODES;         // G
  float* bnsums = gcnt   + (size_t)N_GRAPHS;        // 128
  float* pooled = bnsums + 128;                     // G*64
  int* nodeMaxBits = (int*)(pooled + (size_t)N_GRAPHS * 64);  // N

  const int mtiles = N_NODES / 16;  // 3125 (exact)
  const int etiles = N_EDGES / 16;  // 100000 (exact)

  // in-degree (same every layer) + graph node counts
  hipMemsetAsync(cnt, 0, (size_t)(N_NODES + N_GRAPHS) * sizeof(float), stream);
  degree_kernel<<<(N_EDGES + 255) / 256, 256, 0, stream>>>(dstIdx, cnt, N_EDGES);
  gcount_kernel<<<(N_NODES + 255) / 256, 256, 0, stream>>>(batch, gcnt, N_NODES);

  // pre FC
  pre_fc_kernel<<<(mtiles + WPB - 1) / WPB, TPB, 0, stream>>>(x, pre_w, pre_b, h, mtiles);

  const int nElem  = N_NODES * 64;
  const int eBlocks = (etiles + WPB - 1) / WPB;

  for (int L = 0; L < N_LAYERS; ++L) {
    const float* Wx    = conv_W + (size_t)L * 114 * 64;       // rows 0..63  (node part)
    const float* We    = Wx + 64 * 64;                        // rows 64..113 (edge part)
    const float* attL  = conv_att  + (size_t)L * 128;
    const float* biasL = conv_bias + (size_t)L * 64;

    xw_kernel<<<(mtiles + WPB - 1) / WPB, TPB, 0, stream>>>(h, Wx, xW, mtiles);

    hipMemsetAsync(nodeMaxBits, 0, (size_t)N_NODES * sizeof(int), stream);
    hipMemsetAsync(denom, 0, (size_t)N_NODES * sizeof(float), stream);
    hipMemsetAsync(accum, 0, (size_t)nElem * sizeof(float), stream);
    hipMemsetAsync(bnsums, 0, 128 * sizeof(float), stream);

    edge_attn_kernel<<<eBlocks, TPB, 0, stream>>>(edge_attr, We, xW, srcIdx, dstIdx,
                                                  attL, alpha, nodeMaxBits, etiles);
    edge_exp_kernel<<<(N_EDGES + 255) / 256, 256, 0, stream>>>(alpha, dstIdx, nodeMaxBits,
                                                               eexp, denom, N_EDGES);
    edge_scatter_kernel<<<eBlocks, TPB, 0, stream>>>(edge_attr, We, xW, srcIdx, dstIdx,
                                                     eexp, denom, accum, etiles);
    mean_bias_kernel<<<(nElem + 255) / 256, 256, 0, stream>>>(accum, cnt, biasL, N_NODES);
    bn_stats_kernel<<<512, 256, 0, stream>>>(accum, bnsums, N_NODES);
    bn_apply_kernel<<<(nElem + 255) / 256, 256, 0, stream>>>(accum, bnsums,
                                                             bn_g + (size_t)L * 64,
                                                             bn_b + (size_t)L * 64,
                                                             h, N_NODES);
  }

  hipMemsetAsync(pooled, 0, (size_t)N_GRAPHS * 64 * sizeof(float), stream);
  pool_kernel<<<(nElem + 255) / 256, 256, 0, stream>>>(h, batch, pooled, N_NODES);
  head_kernel<<<N_GRAPHS, 64, 0, stream>>>(pooled, gcnt, post_w, post_b,
                                           out_w, out_b, (float*)d_out);
}